// DeformableWindowAttention_18245021073453
// MI455X (gfx1250) — compile-verified
//
#include <hip/hip_runtime.h>

typedef __attribute__((ext_vector_type(2))) float v2f;
typedef __attribute__((ext_vector_type(8))) float v8f;

#define B_    256
#define N_    49
#define C_    128
#define H_    4
#define HD_   32
#define K2_   49
#define QKV_D 384
#define OFF_D 392
#define OFF_P 448          /* 392 padded to 7 quads of 64 cols */
#define M_    (B_ * N_)    /* 12544 rows, multiple of 32 */

// ---------------------------------------------------------------------------
// Register-blocked GEMM with bias:  Y[M x Nd] = X[M x 128] @ W[Np x 128]^T + b
// Each wave computes a 32x64 output patch = 2 (M) x 4 (N) WMMA tiles.
// Per k-step: 2 A loads + 4 B loads feed 8 V_WMMA_F32_16X16X4_F32
// (load:wmma = 0.75 vs 2.0 unblocked; ~2.7x less operand traffic from L2).
// K=128 compile-time -> fully unrolled: 256 WMMAs per wave, zero divergence
// in the loop (W/bias zero-padded to Np = 64-quad multiple). Ragged Nd only
// affects the store guard.
//
// A layout (ISA 7.12.2, 32-bit A 16x4): lanes 0-15 hold K=0,1; lanes 16-31
// hold K=2,3 for rows M=lane&15. B mirrored: lane half selects the K pair,
// l15 selects the column. C/D: VGPR r -> (M = r + 8*half, N = l15).
// ---------------------------------------------------------------------------
__global__ __launch_bounds__(128)
void gemm_bias_wmma(const float* __restrict__ X,
                    const float* __restrict__ W,
                    const float* __restrict__ bias,
                    float* __restrict__ Y,
                    int Nd) {
  constexpr int KD = 128;

  const int lane = threadIdx.x & 31;
  const int wave = threadIdx.x >> 5;
  const int half = lane >> 4;   // 0 | 1
  const int l15  = lane & 15;

  const int m0 = blockIdx.x * 32;                 // 2 M-tiles per wave
  const int n0 = (blockIdx.y * 4 + wave) * 64;    // 4 N-tiles per wave
  if (n0 >= Nd) return;              // wave-uniform: EXEC stays all-1s inside

  const float* xrow0 = X + (size_t)(m0 + l15) * KD + 2 * half;
  const float* xrow1 = xrow0 + (size_t)16 * KD;
  const float* wrow0 = W + (size_t)(n0 + l15) * KD + 2 * half;  // padded-valid
  const float* wrow1 = wrow0 + (size_t)16 * KD;
  const float* wrow2 = wrow0 + (size_t)32 * KD;
  const float* wrow3 = wrow0 + (size_t)48 * KD;

  // Pull operand rows toward the caches (global_prefetch_b8).
  __builtin_prefetch(xrow0, 0, 3);
  __builtin_prefetch(xrow1, 0, 3);
  __builtin_prefetch(wrow0, 0, 3);
  __builtin_prefetch(wrow2, 0, 3);

  v8f acc[2][4] = {};
#pragma unroll
  for (int k = 0; k < KD; k += 4) {
    const v2f a0 = *(const v2f*)(xrow0 + k);
    const v2f a1 = *(const v2f*)(xrow1 + k);
    const v2f b0 = *(const v2f*)(wrow0 + k);
    const v2f b1 = *(const v2f*)(wrow1 + k);
    const v2f b2 = *(const v2f*)(wrow2 + k);
    const v2f b3 = *(const v2f*)(wrow3 + k);

    acc[0][0] = __builtin_amdgcn_wmma_f32_16x16x4_f32(false, a0, false, b0,
                                                      (short)0, acc[0][0], false, false);
    acc[0][1] = __builtin_amdgcn_wmma_f32_16x16x4_f32(false, a0, false, b1,
                                                      (short)0, acc[0][1], false, false);
    acc[0][2] = __builtin_amdgcn_wmma_f32_16x16x4_f32(false, a0, false, b2,
                                                      (short)0, acc[0][2], false, false);
    acc[0][3] = __builtin_amdgcn_wmma_f32_16x16x4_f32(false, a0, false, b3,
                                                      (short)0, acc[0][3], false, false);
    acc[1][0] = __builtin_amdgcn_wmma_f32_16x16x4_f32(false, a1, false, b0,
                                                      (short)0, acc[1][0], false, false);
    acc[1][1] = __builtin_amdgcn_wmma_f32_16x16x4_f32(false, a1, false, b1,
                                                      (short)0, acc[1][1], false, false);
    acc[1][2] = __builtin_amdgcn_wmma_f32_16x16x4_f32(false, a1, false, b2,
                                                      (short)0, acc[1][2], false, false);
    acc[1][3] = __builtin_amdgcn_wmma_f32_16x16x4_f32(false, a1, false, b3,
                                                      (short)0, acc[1][3], false, false);
  }

#pragma unroll
  for (int j = 0; j < 4; ++j) {
    const int col = n0 + j * 16 + l15;
    if (col < Nd) {                  // only the OFF gemm's ragged tail masks
      const float bval = bias[col];
#pragma unroll
      for (int i = 0; i < 2; ++i) {
#pragma unroll
        for (int r = 0; r < 8; ++r) {
          const int m = m0 + i * 16 + half * 8 + r;
          Y[(size_t)m * Nd + col] = acc[i][j][r] + bval;
        }
      }
    }
  }
}

// ---------------------------------------------------------------------------
// Zero-pad w_off (392x128 -> 448x128) and b_off (392 -> 448) into workspace
// so the GEMM inner loop never needs a predicated load.
// ---------------------------------------------------------------------------
__global__ void pad_off_weights(const float* __restrict__ wsrc,
                                const float* __restrict__ bsrc,
                                float* __restrict__ wdst,
                                float* __restrict__ bdst) {
  const int t = blockIdx.x * 256 + threadIdx.x;
  if (t < OFF_P * C_) {
    const int r = t / C_;
    wdst[t] = (r < OFF_D) ? wsrc[t] : 0.0f;
  }
  if (t < OFF_P) {
    bdst[t] = (t < OFF_D) ? bsrc[t] : 0.0f;
  }
}

// ---------------------------------------------------------------------------
// Precompute relative-position-bias table: btab[h*49+j] = rpb[h, dy+6, dx+6]
// ---------------------------------------------------------------------------
__global__ void build_bias_tab(const float* __restrict__ rpb,
                               float* __restrict__ btab) {
  const int t = threadIdx.x;
  if (t >= H_ * K2_) return;
  const int h  = t / K2_;
  const int j  = t % K2_;
  const int dy = j / 7 - 3;
  const int dx = j % 7 - 3;
  btab[t] = rpb[h * 169 + (dy + 6) * 13 + (dx + 6)];  // (2K-1)=13
}

// ---------------------------------------------------------------------------
// Deformable sampling + attention. One wave32 per (b, h, n); lane = channel d
// of the 32-wide head dim. Sampled v and the 49 scores are staged in LDS
// (~26 KB / 128-thread block, far under the 320 KB WGP budget), then an
// exact softmax + attn@v contraction runs from registers/LDS.
// ---------------------------------------------------------------------------
__global__ __launch_bounds__(128)
void deform_attn(const float* __restrict__ qkv,
                 const float* __restrict__ off,
                 const float* __restrict__ btab,
                 float* __restrict__ out) {
  __shared__ float vs[4][K2_][HD_];
  __shared__ float sc[4][K2_ + 1];

  const int lane = threadIdx.x & 31;
  const int wave = threadIdx.x >> 5;
  const int w    = blockIdx.x * 4 + wave;       // 0 .. B*H*N-1
  const int b    = w / (H_ * N_);
  const int rem  = w - b * (H_ * N_);
  const int h    = rem / N_;
  const int n    = rem - h * N_;
  const int row  = b * N_ + n;

  const float scale = 0.17677669529663689f;     // 1/sqrt(32)
  const float qd = qkv[(size_t)row * QKV_D + h * HD_ + lane] * scale;

  const float* kbase = qkv + (size_t)(b * N_) * QKV_D + C_     + h * HD_ + lane;
  const float* vbase = qkv + (size_t)(b * N_) * QKV_D + 2 * C_ + h * HD_ + lane;
  const float* orow  = off + (size_t)row * OFF_D + h * (K2_ * 2);

  const int iy = n / 7;
  const int ix = n - iy * 7;

  for (int j = 0; j < K2_; ++j) {
    const float oy = orow[2 * j + 0];
    const float ox = orow[2 * j + 1];
    const int jy = j / 7;
    const int dy = jy - 3;
    const int dx = (j - jy * 7) - 3;

    float py = fminf(fmaxf((float)(iy + dy) + oy, 0.0f), 6.0f);
    float px = fminf(fmaxf((float)(ix + dx) + ox, 0.0f), 6.0f);
    const float y0f = floorf(py), x0f = floorf(px);
    const float wy = py - y0f,   wx = px - x0f;
    const int y0 = (int)y0f,     x0 = (int)x0f;
    const int y1 = min(y0 + 1, 6), x1 = min(x0 + 1, 6);

    const int i00 = (y0 * 7 + x0) * QKV_D;
    const int i01 = (y0 * 7 + x1) * QKV_D;
    const int i10 = (y1 * 7 + x0) * QKV_D;
    const int i11 = (y1 * 7 + x1) * QKV_D;

    const float w00 = (1.0f - wy) * (1.0f - wx);
    const float w01 = (1.0f - wy) * wx;
    const float w10 = wy * (1.0f - wx);
    const float w11 = wy * wx;

    const float ksamp = kbase[i00] * w00 + kbase[i01] * w01 +
                        kbase[i10] * w10 + kbase[i11] * w11;
    const float vsamp = vbase[i00] * w00 + vbase[i01] * w01 +
                        vbase[i10] * w10 + vbase[i11] * w11;

    vs[wave][j][lane] = vsamp;

    float p = qd * ksamp;                       // wave32 tree reduction
#pragma unroll
    for (int o = 16; o > 0; o >>= 1) p += __shfl_xor(p, o, 32);
    if (lane == 0) sc[wave][j] = p + btab[h * K2_ + j];
  }
  __syncthreads();

  float mx = -3.4e38f;
  for (int j = 0; j < K2_; ++j) mx = fmaxf(mx, sc[wave][j]);

  float sum = 0.0f, acc = 0.0f;
  for (int j = 0; j < K2_; ++j) {
    const float e = __expf(sc[wave][j] - mx);
    sum += e;
    acc += e * vs[wave][j][lane];
  }
  out[(size_t)row * C_ + h * HD_ + lane] = acc / sum;
}

// ---------------------------------------------------------------------------
// Launch: pad weights + bias table -> QKV GEMM -> offset GEMM -> attention
//         -> proj GEMM.
// Workspace layout (all chunks naturally 256B aligned):
//   qkv      : M*384 f32 (19.27 MB)
//   off      : M*392 f32 (19.67 MB)
//   aout     : M*128 f32 ( 6.42 MB)
//   btab     : 256  f32
//   woff_pad : 448*128 f32 (229 KB)
//   boff_pad : 448 f32
// ---------------------------------------------------------------------------
extern "C" void kernel_launch(void* const* d_in, const int* in_sizes, int n_in,
                              void* d_out, int out_size, void* d_ws, size_t ws_size,
                              hipStream_t stream) {
  const float* x      = (const float*)d_in[0];
  const float* w_qkv  = (const float*)d_in[1];
  const float* b_qkv  = (const float*)d_in[2];
  const float* w_off  = (const float*)d_in[3];
  const float* b_off  = (const float*)d_in[4];
  const float* rpb    = (const float*)d_in[5];
  const float* w_proj = (const float*)d_in[6];
  const float* b_proj = (const float*)d_in[7];
  float* out = (float*)d_out;

  char* ws = (char*)d_ws;
  const size_t qkv_bytes  = (size_t)M_ * QKV_D * sizeof(float);
  const size_t off_bytes  = (size_t)M_ * OFF_D * sizeof(float);
  const size_t aout_bytes = (size_t)M_ * C_ * sizeof(float);
  const size_t btab_bytes = 256 * sizeof(float);
  const size_t wpad_bytes = (size_t)OFF_P * C_ * sizeof(float);

  float* qkv      = (float*)ws;
  float* offb     = (float*)(ws + qkv_bytes);
  float* aout     = (float*)(ws + qkv_bytes + off_bytes);
  float* btab     = (float*)(ws + qkv_bytes + off_bytes + aout_bytes);
  float* woff_pad = (float*)(ws + qkv_bytes + off_bytes + aout_bytes + btab_bytes);
  float* boff_pad = (float*)(ws + qkv_bytes + off_bytes + aout_bytes + btab_bytes + wpad_bytes);

  pad_off_weights<<<(OFF_P * C_ + 255) / 256, 256, 0, stream>>>(w_off, b_off,
                                                                woff_pad, boff_pad);
  build_bias_tab<<<1, 256, 0, stream>>>(rpb, btab);

  {
    dim3 grid(M_ / 32, 2);   // 384 cols = 6 quads of 64; 8 wave-slots, 2 idle
    gemm_bias_wmma<<<grid, 128, 0, stream>>>(x, w_qkv, b_qkv, qkv, QKV_D);
  }
  {
    dim3 grid(M_ / 32, 2);   // 448 padded cols = 7 quads; 8 wave-slots, 1 idle
    gemm_bias_wmma<<<grid, 128, 0, stream>>>(x, woff_pad, boff_pad, offb, OFF_D);
  }

  deform_attn<<<M_, 128, 0, stream>>>(qkv, offb, btab, aout);

  {
    dim3 grid(M_ / 32, 1);   // 128 cols = 2 quads; 4 wave-slots, 2 idle
    gemm_bias_wmma<<<grid, 128, 0, stream>>>(aout, w_proj, b_proj, out, C_);
  }
}